// MetaModel_78460462563658
// MI455X (gfx1250) — compile-verified
//
#include <hip/hip_runtime.h>
#include <hip/hip_bf16.h>

typedef __attribute__((ext_vector_type(2))) float v2f;
typedef __attribute__((ext_vector_type(8))) float v8f;

#define N_STEPS     1000
#define LR_INNER    0.1f
#define LOG_CLAMP_F (-100.0f)

// 16-lane-row xor-butterfly step in the VALU via DPP16 (no LDS round-trip).
// ctrl: 0xB1 = quad_perm[1,0,3,2] (xor 1), 0x4E = quad_perm[2,3,0,1] (xor 2),
//       0x141 = row_half_mirror (xor 4), 0x140 = row_mirror (xor 8).
#define DPP_XADD(v, ctrl)                                                     \
    ((v) + __int_as_float(__builtin_amdgcn_update_dpp(                        \
               0, __float_as_int(v), (ctrl), 0xf, 0xf, true)))

// dL/dz2 for clamped BCE + sigmoid with p/(1-p) cancelled symbolically:
//   dz2 = -1/4 * ( y*mp*(1-p) - (1-y)*mq*p ),  mp/mq = clip() gradient masks.
__device__ __forceinline__ float bce_dz2(float p, float y, float lp, float lq) {
    float tp = (lp >= LOG_CLAMP_F) ? (y * (1.0f - p)) : 0.0f;
    float tq = (lq >= LOG_CLAMP_F) ? ((1.0f - y) * p) : 0.0f;
    return -0.25f * (tp - tq);
}

__device__ __forceinline__ float sigm(float z) {
    return __builtin_amdgcn_rcpf(1.0f + __expf(-z));
}

// One wave32 runs the whole 1000-step training loop; all state in VGPRs.
// Layer-1 matmul x@w1^T is a zero-padded V_WMMA_F32_16X16X4_F32 whose D-matrix
// layout (lane n = column n, VGPR m = sample row m) matches per-lane ownership.
__global__ __launch_bounds__(32, 1)
void xor_mlp_meta_train(const float* __restrict__ xg,
                        const float* __restrict__ yg,
                        const float* __restrict__ w1g,
                        const float* __restrict__ b1g,
                        const float* __restrict__ w2g,
                        const float* __restrict__ b2g,
                        float* __restrict__ out) {
    const int   lane = threadIdx.x;       // 0..31, single wave
    const int   j    = lane & 15;         // hidden unit owned by this lane
    const float m16  = (lane < 16) ? 1.0f : 0.0f;

    // ---- load the tiny problem into registers (every lane) ----
    const float x00 = xg[0], x01 = xg[1];
    const float x10 = xg[2], x11 = xg[3];
    const float x20 = xg[4], x21 = xg[5];
    const float x30 = xg[6], x31 = xg[7];
    const float y0 = yg[0], y1 = yg[1], y2 = yg[2], y3 = yg[3];

    // Upper half-wave (padded K=2,3 rows of the WMMA B operand) zeroed at
    // init; updates are masked, so these stay exactly 0 forever.
    float w1j0 = w1g[2 * j + 0] * m16;   // w1[j][0] -> WMMA B vgpr0
    float w1j1 = w1g[2 * j + 1] * m16;   // w1[j][1] -> WMMA B vgpr1
    float b1j  = b1g[j];
    float w2j  = w2g[j];
    float b2   = b2g[0];

    // WMMA A operand: x zero-padded to 16x4 (4 live rows, K=2 live cols).
    // A layout: vgpr0 = {M=lane,K=0 | M=lane-16,K=2}; vgpr1 = {K=1 | K=3}.
    const int   xr  = lane & 3;
    const float axl = xg[2 * xr + 0];
    const float ayl = xg[2 * xr + 1];
    v2f A;
    A.x = (lane < 4) ? axl : 0.0f;
    A.y = (lane < 4) ? ayl : 0.0f;

    float p0 = 0.f, p1 = 0.f, p2 = 0.f, p3 = 0.f;

    for (int it = 0; it <= N_STEPS; ++it) {
        // ---- forward layer 1 on the matrix pipe ----
        v2f B;
        B.x = w1j0;
        B.y = w1j1;
        v8f C = {0.f, 0.f, 0.f, 0.f, 0.f, 0.f, 0.f, 0.f};
        C = __builtin_amdgcn_wmma_f32_16x16x4_f32(false, A, false, B,
                                                  (short)0, C, false, false);
        // z1 rows 0..3 land in C[0..3]; lane n (n<16) holds column n.
        float h0 = sigm(C[0] + b1j);
        float h1 = sigm(C[1] + b1j);
        float h2 = sigm(C[2] + b1j);
        float h3 = sigm(C[3] + b1j);

        // ---- layer 2: z2[m] = sum_j h[m][j]*w2[j], DPP16 xor-butterfly ----
        float t0 = h0 * w2j, t1 = h1 * w2j, t2 = h2 * w2j, t3 = h3 * w2j;
        t0 = DPP_XADD(t0, 0xB1);  t1 = DPP_XADD(t1, 0xB1);
        t2 = DPP_XADD(t2, 0xB1);  t3 = DPP_XADD(t3, 0xB1);
        t0 = DPP_XADD(t0, 0x4E);  t1 = DPP_XADD(t1, 0x4E);
        t2 = DPP_XADD(t2, 0x4E);  t3 = DPP_XADD(t3, 0x4E);
        t0 = DPP_XADD(t0, 0x141); t1 = DPP_XADD(t1, 0x141);
        t2 = DPP_XADD(t2, 0x141); t3 = DPP_XADD(t3, 0x141);
        t0 = DPP_XADD(t0, 0x140); t1 = DPP_XADD(t1, 0x140);
        t2 = DPP_XADD(t2, 0x140); t3 = DPP_XADD(t3, 0x140);

        // ---- layer-2 sigmoid + both clamped logs from ONE exp per sample:
        //   u = e^{-z}; p = 1/(1+u); log p = -log(1+u); log(1-p) = -z - log(1+u)
        float z20 = t0 + b2, z21 = t1 + b2, z22 = t2 + b2, z23 = t3 + b2;
        float u0 = __expf(-z20), u1 = __expf(-z21);
        float u2 = __expf(-z22), u3 = __expf(-z23);
        float d0 = 1.0f + u0, d1 = 1.0f + u1, d2 = 1.0f + u2, d3 = 1.0f + u3;
        p0 = __builtin_amdgcn_rcpf(d0);
        p1 = __builtin_amdgcn_rcpf(d1);
        p2 = __builtin_amdgcn_rcpf(d2);
        p3 = __builtin_amdgcn_rcpf(d3);

        if (it == N_STEPS) break;   // last pass: forward only -> final_output

        float s0 = __logf(d0), s1 = __logf(d1), s2 = __logf(d2), s3 = __logf(d3);
        float lp0 = -s0, lq0 = -z20 - s0;
        float lp1 = -s1, lq1 = -z21 - s1;
        float lp2 = -s2, lq2 = -z22 - s2;
        float lp3 = -s3, lq3 = -z23 - s3;

        // ---- clamped BCE loss (recorded pre-update, like value_and_grad) ----
        float loss = -0.25f *
            (y0 * fmaxf(lp0, LOG_CLAMP_F) + (1.0f - y0) * fmaxf(lq0, LOG_CLAMP_F) +
             y1 * fmaxf(lp1, LOG_CLAMP_F) + (1.0f - y1) * fmaxf(lq1, LOG_CLAMP_F) +
             y2 * fmaxf(lp2, LOG_CLAMP_F) + (1.0f - y2) * fmaxf(lq2, LOG_CLAMP_F) +
             y3 * fmaxf(lp3, LOG_CLAMP_F) + (1.0f - y3) * fmaxf(lq3, LOG_CLAMP_F));
        if (lane == 0) out[4 + it] = loss;

        // ---- backward (all per-lane; no cross-lane traffic needed) ----
        float dz2_0 = bce_dz2(p0, y0, lp0, lq0);
        float dz2_1 = bce_dz2(p1, y1, lp1, lq1);
        float dz2_2 = bce_dz2(p2, y2, lp2, lq2);
        float dz2_3 = bce_dz2(p3, y3, lp3, lq3);

        float gw2 = dz2_0 * h0 + dz2_1 * h1 + dz2_2 * h2 + dz2_3 * h3; // grad_w2[j]
        float gb2 = dz2_0 + dz2_1 + dz2_2 + dz2_3;                     // replicated

        float dz1_0 = dz2_0 * w2j * h0 * (1.0f - h0);
        float dz1_1 = dz2_1 * w2j * h1 * (1.0f - h1);
        float dz1_2 = dz2_2 * w2j * h2 * (1.0f - h2);
        float dz1_3 = dz2_3 * w2j * h3 * (1.0f - h3);

        float gw10 = dz1_0 * x00 + dz1_1 * x10 + dz1_2 * x20 + dz1_3 * x30;
        float gw11 = dz1_0 * x01 + dz1_1 * x11 + dz1_2 * x21 + dz1_3 * x31;
        float gb1  = dz1_0 + dz1_1 + dz1_2 + dz1_3;

        // ---- SGD update; upper half-wave masked so padded B rows stay 0 ----
        w1j0 -= LR_INNER * gw10 * m16;
        w1j1 -= LR_INNER * gw11 * m16;
        b1j  -= LR_INNER * gb1  * m16;
        w2j  -= LR_INNER * gw2  * m16;
        b2   -= LR_INNER * gb2  * m16;
    }

    if (lane == 0) {   // final forward output [4,1]
        out[0] = p0; out[1] = p1; out[2] = p2; out[3] = p3;
    }
}

extern "C" void kernel_launch(void* const* d_in, const int* in_sizes, int n_in,
                              void* d_out, int out_size, void* d_ws, size_t ws_size,
                              hipStream_t stream) {
    const float* x  = (const float*)d_in[0];
    const float* y  = (const float*)d_in[1];
    const float* w1 = (const float*)d_in[2];
    const float* b1 = (const float*)d_in[3];
    const float* w2 = (const float*)d_in[4];
    const float* b2 = (const float*)d_in[5];
    float* out = (float*)d_out;
    // Strictly sequential training chain: exactly one wave32 does everything.
    xor_mlp_meta_train<<<dim3(1), dim3(32), 0, stream>>>(x, y, w1, b1, w2, b2, out);
}